// ThresholdSigmoidMask_67035849556096
// MI455X (gfx1250) — compile-verified
//
#include <hip/hip_runtime.h>
#include <stdint.h>

// ThresholdSigmoidMask for MI455X (gfx1250, wave32).
// Bit-exact JAX threefry2x32 RNG reproduction; WMMA-based wave reductions;
// 128-bit vector loads/stores; ds_swizzle cross-half exchange.

#define B_COUNT   8
#define N_PER_B   (2048 * 2048)     // 4194304 elements per batch
#define H_HALF    (N_PER_B / 2)     // 2097152 threefry counter pairs
#define T_MAXA    4                 // attempt budget (P(reject) ~ 4e-5 per attempt)
#define BLOCKS_PB 512
#define NTHREADS  256

typedef __attribute__((ext_vector_type(2))) float v2f;
typedef __attribute__((ext_vector_type(4))) float v4f;
typedef __attribute__((ext_vector_type(8))) float v8f;

struct Scratch {
  uint32_t subkeys[B_COUNT][T_MAXA][2];   // per-batch, per-attempt uniform() keys
  uint32_t counts[B_COUNT][T_MAXA];       // popcount(x > prob_t), exact u32 atomics
  uint32_t chosen[B_COUNT];               // accepted attempt index
  float    tsum[B_COUNT];                 // sum(x) per batch
  float    partials[B_COUNT][BLOCKS_PB];  // stage-1 block sums (fixed-order stage 2)
};

// ---- JAX threefry2x32 (20 rounds), exact key schedule ----------------------
__device__ __forceinline__ void threefry2x32(uint32_t k0, uint32_t k1,
                                             uint32_t c0, uint32_t c1,
                                             uint32_t& o0, uint32_t& o1) {
  const uint32_t ks2 = k0 ^ k1 ^ 0x1BD11BDAu;
  uint32_t x0 = c0 + k0, x1 = c1 + k1;
#define TF_ROUND(r) { x0 += x1; x1 = (x1 << (r)) | (x1 >> (32 - (r))); x1 ^= x0; }
  TF_ROUND(13) TF_ROUND(15) TF_ROUND(26) TF_ROUND(6)
  x0 += k1;  x1 += ks2 + 1u;
  TF_ROUND(17) TF_ROUND(29) TF_ROUND(16) TF_ROUND(24)
  x0 += ks2; x1 += k0 + 2u;
  TF_ROUND(13) TF_ROUND(15) TF_ROUND(26) TF_ROUND(6)
  x0 += k0;  x1 += k1 + 3u;
  TF_ROUND(17) TF_ROUND(29) TF_ROUND(16) TF_ROUND(24)
  x0 += k1;  x1 += ks2 + 4u;
  TF_ROUND(13) TF_ROUND(15) TF_ROUND(26) TF_ROUND(6)
  x0 += ks2; x1 += k0 + 5u;
#undef TF_ROUND
  o0 = x0; o1 = x1;
}

// jax.random.uniform bit manipulation: (bits>>9)|0x3f800000, bitcast, -1.0
__device__ __forceinline__ float bits_to_uniform(uint32_t bits) {
  return __uint_as_float((bits >> 9) | 0x3f800000u) - 1.0f;  // in [0,1)
}

// ds_swizzle_b32 bitmask mode (group-of-32: and=0x10, or=0, xor=0x1F):
// result[lane] = src[(lane & 16) ^ 31]; equals src[lane ^ 16] when each half is uniform.  EXEC all-ones at call sites.
__device__ __forceinline__ float swizzle_xor16(float v) {
  return __int_as_float(__builtin_amdgcn_ds_swizzle(__float_as_int(v), 0x7C10));
}

// ---- Exact 32-lane sum via V_WMMA_F32_16X16X4_F32 --------------------------
// A: lane L -> {v, 0} puts v at A[L%16][K = L<16 ? 0 : 2]; B = ones, so
// D[m][n] = v_m + v_{m+16}.  Lanes 0-15 hold rows 0..7 in c[0..7], lanes
// 16-31 hold rows 8..15; sum 8 accumulator VGPRs + one xor-16 swizzle gives
// the full wave sum in every lane (exact for integer-valued f32 inputs).
// EXEC must be all ones at the call site.
__device__ __forceinline__ float wave_sum_wmma(float v) {
  v2f a; a.x = v;    a.y = 0.0f;
  v2f b; b.x = 1.0f; b.y = 1.0f;
  v8f c = {};
  c = __builtin_amdgcn_wmma_f32_16x16x4_f32(
      /*neg_a=*/false, a, /*neg_b=*/false, b,
      /*c_mod=*/(short)0, c, /*reuse_a=*/false, /*reuse_b=*/false);
  float s = ((c[0] + c[1]) + (c[2] + c[3])) + ((c[4] + c[5]) + (c[6] + c[7]));
  s += swizzle_xor16(s);
  return s;
}

__device__ __forceinline__ float block_sum(float wsum, float* lds8) {
  const int wave = threadIdx.x >> 5, lane = threadIdx.x & 31;
  if (lane == 0) lds8[wave] = wsum;
  __syncthreads();
  float s = 0.0f;
  if (threadIdx.x == 0) {
    #pragma unroll
    for (int w = 0; w < NTHREADS / 32; ++w) s += lds8[w];  // fixed order
  }
  __syncthreads();
  return s;  // valid in thread 0
}

// ---- Kernel 0: derive per-batch / per-attempt keys (tiny) ------------------
// keys = split(key(42), 8): key_b = (f(2b), f(2b+1)),
//   f(i) = i<8 ? word0(threefry((0,42),(i,i+8))) : word1(threefry((0,42),(i-8,i)))
// split(k) -> knew = (w0(0,2), w0(1,3)), sub = (w1(0,2), w1(1,3))
__global__ void k_keys(Scratch* ws) {
  const int b = threadIdx.x;
  if (b >= B_COUNT) return;
  uint32_t k0, k1;
  {
    uint32_t a, c;
    int i = 2 * b;
    if (i < 8) { threefry2x32(0u, 42u, (uint32_t)i, (uint32_t)(i + 8), a, c); k0 = a; }
    else       { threefry2x32(0u, 42u, (uint32_t)(i - 8), (uint32_t)i, a, c); k0 = c; }
    i = 2 * b + 1;
    if (i < 8) { threefry2x32(0u, 42u, (uint32_t)i, (uint32_t)(i + 8), a, c); k1 = a; }
    else       { threefry2x32(0u, 42u, (uint32_t)(i - 8), (uint32_t)i, a, c); k1 = c; }
  }
  for (int t = 0; t < T_MAXA; ++t) {
    uint32_t a0, a1, b0, b1;
    threefry2x32(k0, k1, 0u, 2u, a0, a1);
    threefry2x32(k0, k1, 1u, 3u, b0, b1);
    ws->subkeys[b][t][0] = a1;   // sub_t
    ws->subkeys[b][t][1] = b1;
    k0 = a0; k1 = b0;            // knew
  }
}

// ---- Kernel 1: per-batch sum(x), stage 1 (b128 loads + WMMA wave sums) -----
__global__ void k_batch_sum(const float* __restrict__ x, Scratch* ws) {
  const int b = blockIdx.y;
  const v4f* xb4 = (const v4f*)(x + (size_t)b * N_PER_B);
  const int base = blockIdx.x * 2048;          // 2048 float4 = 8192 floats/block
  v4f acc = {0.f, 0.f, 0.f, 0.f};
  #pragma unroll
  for (int i = 0; i < 8; ++i) {
    const int q = base + i * NTHREADS + threadIdx.x;
    __builtin_prefetch(&xb4[q + 2 * NTHREADS], 0, 0);   // global_prefetch_b8
    const v4f v = xb4[q];                               // global_load_b128
    acc.x += v.x; acc.y += v.y; acc.z += v.z; acc.w += v.w;
  }
  float wsum = wave_sum_wmma((acc.x + acc.y) + (acc.z + acc.w));
  __shared__ float lds8[NTHREADS / 32];
  float bsum = block_sum(wsum, lds8);
  if (threadIdx.x == 0) ws->partials[b][blockIdx.x] = bsum;
}

// ---- Kernel 2: stage 2 reduce, fixed-order tree (deterministic) ------------
__global__ void k_reduce_partials(Scratch* ws) {
  const int b = blockIdx.x;
  __shared__ float lds[NTHREADS];
  lds[threadIdx.x] = ws->partials[b][threadIdx.x] +
                     ws->partials[b][threadIdx.x + NTHREADS];
  __syncthreads();
  for (int off = NTHREADS / 2; off > 0; off >>= 1) {
    if (threadIdx.x < off) lds[threadIdx.x] += lds[threadIdx.x + off];
    __syncthreads();
  }
  if (threadIdx.x == 0) ws->tsum[b] = lds[0];
}

// ---- Kernel 3: count(x > prob_t) for all attempts (recompute RNG) ----------
__global__ void k_count(const float* __restrict__ x, Scratch* ws) {
  const int b = blockIdx.y;
  const v4f* xb4 = (const v4f*)(x + (size_t)b * N_PER_B);
  const int base = blockIdx.x * 1024;          // 1024 float4 = 4096 pairs/block
  uint32_t sk0[T_MAXA], sk1[T_MAXA];
  #pragma unroll
  for (int t = 0; t < T_MAXA; ++t) {
    sk0[t] = ws->subkeys[b][t][0];
    sk1[t] = ws->subkeys[b][t][1];
  }
  int cnt[T_MAXA] = {};
  for (int i = 0; i < 4; ++i) {
    const int q = base + i * NTHREADS + threadIdx.x;    // float4 index, pairs j=4q..4q+3
    __builtin_prefetch(&xb4[q + NTHREADS], 0, 0);
    const v4f xv0 = xb4[q];                             // x[j]      (b128)
    const v4f xv1 = xb4[q + H_HALF / 4];                // x[j+H]    (b128)
    const uint32_t j0 = (uint32_t)q * 4u;
    #pragma unroll
    for (int t = 0; t < T_MAXA; ++t) {
      #pragma unroll
      for (int e = 0; e < 4; ++e) {
        uint32_t w0, w1;
        threefry2x32(sk0[t], sk1[t], j0 + e, j0 + e + H_HALF, w0, w1);
        cnt[t] += (xv0[e] > bits_to_uniform(w0)) ? 1 : 0;
        cnt[t] += (xv1[e] > bits_to_uniform(w1)) ? 1 : 0;
      }
    }
  }
  __shared__ float lds8[NTHREADS / 32];
  #pragma unroll
  for (int t = 0; t < T_MAXA; ++t) {
    float wsum = wave_sum_wmma((float)cnt[t]);   // exact: <= 32 per lane
    float bsum = block_sum(wsum, lds8);          // exact: <= 8192
    if (threadIdx.x == 0)
      atomicAdd(&ws->counts[b][t], (uint32_t)(bsum + 0.5f));
    __syncthreads();
  }
}

// ---- Kernel 4: pick first accepted attempt (rejection-loop semantics) ------
__global__ void k_pick(Scratch* ws) {
  const int b = threadIdx.x;
  if (b >= B_COUNT) return;
  const float inv_n  = 1.0f / (float)N_PER_B;
  const float target = ws->tsum[b] * inv_n;
  const float tol    = 1e-3f + 1e-5f * fabsf(target);   // ATOL + RTOL*|target|
  uint32_t c = T_MAXA - 1;
  for (int t = T_MAXA - 1; t >= 0; --t) {
    const float m = (float)ws->counts[b][t] * inv_n;
    if (fabsf(m - target) <= tol) c = (uint32_t)t;       // ends at smallest accepted t
  }
  ws->chosen[b] = c;
}

// ---- Kernel 5: regenerate accepted prob stream, write 0/1 mask (b128) ------
__global__ void k_write(const float* __restrict__ x, const Scratch* __restrict__ ws,
                        float* __restrict__ out) {
  const int b = blockIdx.y;
  const uint32_t t  = ws->chosen[b];
  const uint32_t s0 = ws->subkeys[b][t][0];
  const uint32_t s1 = ws->subkeys[b][t][1];
  const v4f* xb4 = (const v4f*)(x + (size_t)b * N_PER_B);
  v4f* ob4       = (v4f*)(out + (size_t)b * N_PER_B);
  const int base = blockIdx.x * 1024;
  for (int i = 0; i < 4; ++i) {
    const int q = base + i * NTHREADS + threadIdx.x;
    __builtin_prefetch(&xb4[q + NTHREADS], 0, 0);
    const v4f xv0 = xb4[q];
    const v4f xv1 = xb4[q + H_HALF / 4];
    const uint32_t j0 = (uint32_t)q * 4u;
    v4f m0, m1;
    #pragma unroll
    for (int e = 0; e < 4; ++e) {
      uint32_t w0, w1;
      threefry2x32(s0, s1, j0 + e, j0 + e + H_HALF, w0, w1);
      m0[e] = (xv0[e] > bits_to_uniform(w0)) ? 1.0f : 0.0f;
      m1[e] = (xv1[e] > bits_to_uniform(w1)) ? 1.0f : 0.0f;
    }
    ob4[q]               = m0;   // global_store_b128
    ob4[q + H_HALF / 4]  = m1;   // global_store_b128
  }
}

extern "C" void kernel_launch(void* const* d_in, const int* in_sizes, int n_in,
                              void* d_out, int out_size, void* d_ws, size_t ws_size,
                              hipStream_t stream) {
  (void)in_sizes; (void)n_in; (void)out_size; (void)ws_size;
  const float* x = (const float*)d_in[0];          // [8, 2048, 2048] fp32
  float* out     = (float*)d_out;                  // [8, 2048, 2048] fp32 mask
  Scratch* ws    = (Scratch*)d_ws;                 // ~16.8 KB used

  hipMemsetAsync(ws, 0, sizeof(Scratch), stream);  // zero counts (capturable)

  dim3 big(BLOCKS_PB, B_COUNT);
  k_keys<<<1, 32, 0, stream>>>(ws);
  k_batch_sum<<<big, NTHREADS, 0, stream>>>(x, ws);
  k_reduce_partials<<<B_COUNT, NTHREADS, 0, stream>>>(ws);
  k_count<<<big, NTHREADS, 0, stream>>>(x, ws);
  k_pick<<<1, 32, 0, stream>>>(ws);
  k_write<<<big, NTHREADS, 0, stream>>>(x, ws, out);
}